// GCN_68143951118625
// MI455X (gfx1250) — compile-verified
//
#include <hip/hip_runtime.h>

typedef float v2f __attribute__((ext_vector_type(2)));
typedef float v8f __attribute__((ext_vector_type(8)));

// ---------------------------------------------------------------- utilities
__global__ void fill_f32(float* __restrict__ p, float v, long long n) {
    long long i = (long long)blockIdx.x * blockDim.x + threadIdx.x;
    long long stride = (long long)gridDim.x * blockDim.x;
    for (; i < n; i += stride) p[i] = v;
}

// deg[r] starts at 1.0 (self loop); add 1 per incident edge (row side).
__global__ void deg_count(const int* __restrict__ ei_row, float* __restrict__ deg, int E) {
    int e = blockIdx.x * blockDim.x + threadIdx.x;
    if (e < E) unsafeAtomicAdd(&deg[ei_row[e]], 1.0f);
}

__global__ void rsqrt_inplace(float* __restrict__ d, int n) {
    int i = blockIdx.x * blockDim.x + threadIdx.x;
    if (i < n) { float v = d[i]; d[i] = (v > 0.0f) ? rsqrtf(v) : 0.0f; }
}

// Build int32 row/col + normalized edge weight for E real edges + N self loops.
__global__ void build_edges(const int* __restrict__ ei, const float* __restrict__ dis,
                            int E, int N, int* __restrict__ r32, int* __restrict__ c32,
                            float* __restrict__ ew) {
    int e = blockIdx.x * blockDim.x + threadIdx.x;
    int Et = E + N;
    if (e >= Et) return;
    int r, c;
    if (e < E) { r = ei[e]; c = ei[E + e]; }  // edge_index[0][e], edge_index[1][e]
    else       { r = e - E; c = r; }          // self loop
    r32[e] = r; c32[e] = c;
    ew[e] = dis[r] * dis[c];
}

// --------------------------------------------------- dense linear via WMMA f32
// out[M, nColTiles*16] = (RELU? relu(A) : A)[M,K] @ W[cols,K]^T + bias
// One wave computes one 16x16 output tile, K stepped by 4 (V_WMMA_F32_16X16X4_F32).
template <bool RELU>
__global__ void gemm_wmma_f32(const float* __restrict__ A, const float* __restrict__ W,
                              const float* __restrict__ bias, float* __restrict__ out,
                              int nRows, int nColTiles, int K) {
    int wavesPerBlock = blockDim.x >> 5;
    int tile = blockIdx.x * wavesPerBlock + (threadIdx.x >> 5);
    int nRowTiles = (nRows + 15) >> 4;
    if (tile >= nRowTiles * nColTiles) return;       // wave-uniform: EXEC stays all-1s
    int tr = tile / nColTiles;
    int tc = tile - tr * nColTiles;

    int lane  = threadIdx.x & 31;
    int lrow  = lane & 15;                            // M (for A) / N (for B) within tile
    int khalf = (lane >> 4) * 2;                      // lanes 16-31 hold K+2,K+3

    int arow = tr * 16 + lrow;
    if (arow >= nRows) arow = nRows - 1;              // clamp tail (dup loads, guarded store)
    const float* aP = A + (size_t)arow * K + khalf;
    const float* wP = W + (size_t)(tc * 16 + lrow) * K + khalf;   // B[k][n] = W[n][k]

    float bval = bias[tc * 16 + lrow];
    v8f c;
#pragma unroll
    for (int r = 0; r < 8; ++r) c[r] = bval;          // fused bias (broadcast over M)

#pragma unroll 4
    for (int k = 0; k < K; k += 4) {
        v2f a = *(const v2f*)(aP + k);                // A: M=lrow, K = k+khalf+{0,1}
        v2f b = *(const v2f*)(wP + k);                // B: N=lrow, K = k+khalf+{0,1}
        if (RELU) { a.x = fmaxf(a.x, 0.0f); a.y = fmaxf(a.y, 0.0f); }
        c = __builtin_amdgcn_wmma_f32_16x16x4_f32(
                /*neg_a=*/false, a, /*neg_b=*/false, b,
                /*c_mod=*/(short)0, c, /*reuse_a=*/false, /*reuse_b=*/false);
    }

    // C/D layout: VGPR r -> M = r + 8*(lane>=16), N = lane&15
    int ldOut  = nColTiles * 16;
    int rowBase = tr * 16 + (lane >> 4) * 8;
    float* oP = out + (size_t)rowBase * ldOut + tc * 16 + lrow;
#pragma unroll
    for (int r = 0; r < 8; ++r)
        if (rowBase + r < nRows) oP[(size_t)r * ldOut] = c[r];
}

// -------------------------------------------- message passing (scatter-add)
// One wave per edge. F=128: float4 per lane (32*4 = 128), 4 f32 atomics/lane.
__global__ void scatter_add_128(const int* __restrict__ r32, const int* __restrict__ c32,
                                const float* __restrict__ ew, const float* __restrict__ h,
                                float* __restrict__ out, int Et) {
    int e = blockIdx.x * (blockDim.x >> 5) + (threadIdx.x >> 5);
    if (e >= Et) return;
    int lane = threadIdx.x & 31;
    int r = r32[e], c = c32[e];
    float w = ew[e];
    const float4* src = (const float4*)(h + (size_t)c * 128);
    float4 v = src[lane];                              // 128B-coalesced gather row
    float* dst = out + (size_t)r * 128 + lane * 4;
    unsafeAtomicAdd(dst + 0, v.x * w);
    unsafeAtomicAdd(dst + 1, v.y * w);
    unsafeAtomicAdd(dst + 2, v.z * w);
    unsafeAtomicAdd(dst + 3, v.w * w);
}

// F=64: float2 per lane (32*2 = 64), 2 f32 atomics/lane.
__global__ void scatter_add_64(const int* __restrict__ r32, const int* __restrict__ c32,
                               const float* __restrict__ ew, const float* __restrict__ h,
                               float* __restrict__ out, int Et) {
    int e = blockIdx.x * (blockDim.x >> 5) + (threadIdx.x >> 5);
    if (e >= Et) return;
    int lane = threadIdx.x & 31;
    int r = r32[e], c = c32[e];
    float w = ew[e];
    const float2* src = (const float2*)(h + (size_t)c * 64);
    float2 v = src[lane];
    float* dst = out + (size_t)r * 64 + lane * 2;
    unsafeAtomicAdd(dst + 0, v.x * w);
    unsafeAtomicAdd(dst + 1, v.y * w);
}

// --------------------------------------------------------------- launcher
extern "C" void kernel_launch(void* const* d_in, const int* in_sizes, int n_in,
                              void* d_out, int out_size, void* d_ws, size_t ws_size,
                              hipStream_t stream) {
    const float* x  = (const float*)d_in[0];
    const int*   ei = (const int*)  d_in[1];   // [2, E] int32
    const float* W1 = (const float*)d_in[3];   // [HID, IN]
    const float* b1 = (const float*)d_in[4];   // [HID]
    const float* W2 = (const float*)d_in[5];   // [OUT, HID]
    const float* b2 = (const float*)d_in[6];   // [OUT]
    float* out = (float*)d_out;

    const int HID = in_sizes[4];               // 128
    const int OUT = in_sizes[6];               // 64
    const int IN  = in_sizes[3] / HID;         // 128
    const int N   = in_sizes[0] / IN;          // 100000
    const int E   = in_sizes[1] / 2;           // 1600000
    const int Et  = E + N;                     // edges + self loops

    // workspace layout (256B-aligned slices)
    char* ws = (char*)d_ws;
    size_t off = 0;
    auto alloc = [&](size_t bytes) -> void* {
        void* p = ws + off; off += (bytes + 255) & ~(size_t)255; return p;
    };
    float* dis = (float*)alloc((size_t)N * 4);          // deg -> deg^{-1/2}
    float* ew  = (float*)alloc((size_t)Et * 4);
    int*   r32 = (int*)  alloc((size_t)Et * 4);
    int*   c32 = (int*)  alloc((size_t)Et * 4);
    float* h1  = (float*)alloc((size_t)N * HID * 4);    // x@W1^T + b1
    float* a1  = (float*)alloc((size_t)N * HID * 4);    // aggregated layer-1
    float* h2  = (float*)alloc((size_t)N * OUT * 4);    // relu(a1)@W2^T + b2
    (void)ws_size; (void)n_in; (void)out_size;

    // --- normalization data ---
    fill_f32<<<1024, 256, 0, stream>>>(dis, 1.0f, (long long)N);                 // self-loop deg
    fill_f32<<<4096, 256, 0, stream>>>(a1, 0.0f, (long long)N * HID);
    fill_f32<<<4096, 256, 0, stream>>>(out, 0.0f, (long long)N * OUT);
    deg_count<<<(E + 255) / 256, 256, 0, stream>>>(ei, dis, E);
    rsqrt_inplace<<<(N + 255) / 256, 256, 0, stream>>>(dis, N);
    build_edges<<<(Et + 255) / 256, 256, 0, stream>>>(ei, dis, E, N, r32, c32, ew);

    const int rowTiles = (N + 15) / 16;

    // --- layer 1: dense WMMA + edge scatter ---
    {
        int tiles = rowTiles * (HID / 16);
        gemm_wmma_f32<false><<<(tiles + 7) / 8, 256, 0, stream>>>(x, W1, b1, h1, N, HID / 16, IN);
    }
    scatter_add_128<<<(Et + 7) / 8, 256, 0, stream>>>(r32, c32, ew, h1, a1, Et);

    // --- layer 2: fused ReLU into A-load, dense WMMA + edge scatter ---
    {
        int tiles = rowTiles * (OUT / 16);
        gemm_wmma_f32<true><<<(tiles + 7) / 8, 256, 0, stream>>>(a1, W2, b2, h2, N, OUT / 16, HID);
    }
    scatter_add_64<<<(Et + 7) / 8, 256, 0, stream>>>(r32, c32, ew, h2, out, Et);
}